// Gnode_35914516529659
// MI455X (gfx1250) — compile-verified
//
#include <hip/hip_runtime.h>

// ---------------------------------------------------------------------------
// GCN-ODE (RK4, 2 steps) for MI455X / gfx1250.
//
// Math optimization: A_hat @ (H @ W2) == (A_hat @ H) @ W2, so all edge
// gather/scatter runs at feature width 5 (padded to 8), never 128.
// Dense GEMMs use V_WMMA_F32_16X16X4_F32 (full f32 precision; problem is
// memory-bound: all tensors are L2-resident on the 192MB L2).
//
// WMMA f32 16x16x4 VGPR layouts (CDNA5 ISA 7.12.2):
//   A (16x4): lanes 0-15 row M=lane, VGPR{0,1}=K{0,1}; lanes 16-31 K{2,3}
//   B (4x16): lanes 0-15 col N=lane, VGPR{0,1}=K{0,1}; lanes 16-31 K{2,3}
//   C/D (16x16): VGPR j -> row j (lanes 0-15) / row j+8 (lanes 16-31), col=lane&15
// ---------------------------------------------------------------------------

typedef __attribute__((ext_vector_type(2))) float v2f;
typedef __attribute__((ext_vector_type(8))) float v8f;

#define D_FEAT 128

// ---------- degree / normalization -----------------------------------------
__global__ void k_init_deg(float* deg, int N) {
  int i = blockIdx.x * blockDim.x + threadIdx.x;
  if (i < N) deg[i] = 1.0f;  // self-loop contribution
}

__global__ void k_deg_edges(const int* __restrict__ ei, float* __restrict__ deg, int E) {
  int e = blockIdx.x * blockDim.x + threadIdx.x;
  if (e < E) atomicAdd(&deg[ei[E + e]], 1.0f);
}

__global__ void k_dinv(float* dinv, int N) {
  int i = blockIdx.x * blockDim.x + threadIdx.x;
  if (i < N) dinv[i] = rsqrtf(dinv[i]);
}

// ---------- GEMM1: T1[N,5(pad8)] = Y[N,128] @ W1[128,5] ---------------------
__global__ void k_gemm1(const float* __restrict__ Y, const float* __restrict__ W1,
                        float* __restrict__ T1, int N) {
  __shared__ float Bs[128 * 16];  // W1 padded to 16 columns (cols 5..15 = 0)
  for (int idx = threadIdx.x; idx < 128 * 16; idx += blockDim.x) {
    int k = idx >> 4, n = idx & 15;
    Bs[idx] = (n < 5) ? W1[k * 5 + n] : 0.0f;
  }
  __syncthreads();

  int lane = threadIdx.x & 31;
  int wave = threadIdx.x >> 5;
  int r0 = blockIdx.x * 128 + wave * 16;   // 8 waves x 16 rows per block

  int mrow = r0 + (lane & 15);
  if (mrow > N - 1) mrow = N - 1;          // clamp: safe loads, stores guarded
  int koff = (lane < 16) ? 0 : 2;
  int ncol = lane & 15;
  const float* yrow = Y + (size_t)mrow * D_FEAT;

  v8f c = {0.f, 0.f, 0.f, 0.f, 0.f, 0.f, 0.f, 0.f};
#pragma unroll 8
  for (int k0 = 0; k0 < 128; k0 += 4) {
    v2f a, b;
    a.x = yrow[k0 + koff];
    a.y = yrow[k0 + koff + 1];
    b.x = Bs[(k0 + koff) * 16 + ncol];
    b.y = Bs[(k0 + koff + 1) * 16 + ncol];
    c = __builtin_amdgcn_wmma_f32_16x16x4_f32(false, a, false, b, (short)0, c,
                                              false, false);
  }

  int rbase = r0 + ((lane < 16) ? 0 : 8);
#pragma unroll
  for (int j = 0; j < 8; ++j) {
    int row = rbase + j;
    if (row < N && ncol < 8) T1[(size_t)row * 8 + ncol] = c[j];  // cols 5..7 are 0
  }
}

// ---------- self-loop init: out = dinv[i]^2 * in (width 5, zero pads) -------
__global__ void k_selfloop_init(const float* __restrict__ in8, const float* __restrict__ dinv,
                                float* __restrict__ out8, int N) {
  int i = blockIdx.x * blockDim.x + threadIdx.x;
  if (i >= N) return;
  float s = dinv[i];
  s = s * s;
#pragma unroll
  for (int j = 0; j < 5; ++j) out8[(size_t)i * 8 + j] = s * in8[(size_t)i * 8 + j];
  out8[(size_t)i * 8 + 5] = 0.f;
  out8[(size_t)i * 8 + 6] = 0.f;
  out8[(size_t)i * 8 + 7] = 0.f;
}

// ---------- edge scatter: out[dst] += dinv[s]*dinv[d] * in[src] (width 5) ---
__global__ void k_agg_edges(const int* __restrict__ ei, const float* __restrict__ dinv,
                            const float* __restrict__ in8, float* __restrict__ out8, int E) {
  int e = blockIdx.x * blockDim.x + threadIdx.x;
  if (e >= E) return;
  if (e + 8192 < E) {  // gfx1250 global_prefetch_b8 for the edge stream
    __builtin_prefetch(&ei[e + 8192], 0, 1);
    __builtin_prefetch(&ei[E + e + 8192], 0, 1);
  }
  int s = ei[e], d = ei[E + e];
  float nrm = dinv[s] * dinv[d];
  const float* ip = in8 + (size_t)s * 8;
  float* op = out8 + (size_t)d * 8;
#pragma unroll
  for (int j = 0; j < 5; ++j) atomicAdd(op + j, nrm * ip[j]);
}

// ---------- bias + ReLU (width 5) -------------------------------------------
__global__ void k_relu_bias(float* __restrict__ H, const float* __restrict__ b1, int N) {
  int i = blockIdx.x * blockDim.x + threadIdx.x;
  if (i >= N) return;
#pragma unroll
  for (int j = 0; j < 5; ++j) {
    float v = H[(size_t)i * 8 + j] + b1[j];
    H[(size_t)i * 8 + j] = v > 0.f ? v : 0.f;
  }
}

// ---------- GEMM2 fused: k = A2@W2 + b2; ACC (+)= w*k; YP = Y + c*k ---------
__global__ void k_gemm2_fused(const float* __restrict__ A2, const float* __restrict__ W2,
                              const float* __restrict__ b2, const float* __restrict__ Ybase,
                              float* __restrict__ ACC, float* __restrict__ YP,
                              float wAcc, float cStep, int accumulate, int writeYp, int N) {
  __shared__ float Bs[8 * 128];  // W2 padded K: rows 5..7 = 0
  for (int idx = threadIdx.x; idx < 8 * 128; idx += blockDim.x) {
    int k = idx >> 7, n = idx & 127;
    Bs[idx] = (k < 5) ? W2[k * 128 + n] : 0.0f;
  }
  __syncthreads();

  int lane = threadIdx.x & 31;
  int wave = threadIdx.x >> 5;
  int r0 = blockIdx.x * 128 + wave * 16;

  int arow = r0 + (lane & 15);
  if (arow > N - 1) arow = N - 1;
  int koff = (lane < 16) ? 0 : 2;

  v2f a0, a1;  // K=0..3 and K=4..7 fragments of A2 (one load, reused for 8 tiles)
  a0.x = A2[(size_t)arow * 8 + koff];
  a0.y = A2[(size_t)arow * 8 + koff + 1];
  a1.x = A2[(size_t)arow * 8 + 4 + koff];
  a1.y = A2[(size_t)arow * 8 + 4 + koff + 1];

  int nlane = lane & 15;
  int rbase = r0 + ((lane < 16) ? 0 : 8);

  for (int n0 = 0; n0 < 128; n0 += 16) {
    int n = n0 + nlane;
    v2f b0, b1v;
    b0.x = Bs[(0 + koff) * 128 + n];
    b0.y = Bs[(1 + koff) * 128 + n];
    b1v.x = Bs[(4 + koff) * 128 + n];
    b1v.y = Bs[(5 + koff) * 128 + n];

    v8f c = {0.f, 0.f, 0.f, 0.f, 0.f, 0.f, 0.f, 0.f};
    c = __builtin_amdgcn_wmma_f32_16x16x4_f32(false, a0, false, b0, (short)0, c,
                                              false, false);
    c = __builtin_amdgcn_wmma_f32_16x16x4_f32(false, a1, false, b1v, (short)0, c,
                                              false, false);
    float bb = b2[n];
#pragma unroll
    for (int j = 0; j < 8; ++j) {
      int row = rbase + j;
      if (row < N) {
        size_t off = (size_t)row * D_FEAT + n;
        float kv = c[j] + bb;
        float av = accumulate ? (ACC[off] + wAcc * kv) : (wAcc * kv);
        ACC[off] = av;
        if (writeYp) YP[off] = Ybase[off] + cStep * kv;
      }
    }
  }
}

// ---------- RK4 combine: Yout = Yin + scale*ACC -----------------------------
__global__ void k_combine(const float* Yin, const float* ACC, float* Yout,
                          float scale, size_t total) {
  size_t i = (size_t)blockIdx.x * blockDim.x + threadIdx.x;
  if (i < total) Yout[i] = Yin[i] + scale * ACC[i];
}

// ---------------------------------------------------------------------------
extern "C" void kernel_launch(void* const* d_in, const int* in_sizes, int n_in,
                              void* d_out, int out_size, void* d_ws, size_t ws_size,
                              hipStream_t stream) {
  const float* x  = (const float*)d_in[0];
  const int*   ei = (const int*)d_in[1];   // JAX default: int64 request -> int32
  const float* W1 = (const float*)d_in[2];
  const float* b1 = (const float*)d_in[3];
  const float* W2 = (const float*)d_in[4];
  const float* b2 = (const float*)d_in[5];
  float* Yout = (float*)d_out;

  const int N = in_sizes[0] / D_FEAT;
  const int E = in_sizes[1] / 2;
  const size_t total = (size_t)N * D_FEAT;

  // workspace partition (floats)
  float* dinv = (float*)d_ws;           // N
  float* T1   = dinv + N;               // 8N
  float* H    = T1 + (size_t)8 * N;     // 8N
  float* A2   = H + (size_t)8 * N;      // 8N
  float* ACC  = A2 + (size_t)8 * N;     // 128N
  float* YP   = ACC + total;            // 128N

  const int gN    = (N + 255) / 256;
  const int gE    = (E + 255) / 256;
  const int gRows = (N + 127) / 128;    // 8 waves x 16 rows per 256-thread block
  const int gTot  = (int)((total + 255) / 256);

  // symmetric normalization: dinv = rsqrt(deg), deg includes self-loops
  k_init_deg<<<gN, 256, 0, stream>>>(dinv, N);
  k_deg_edges<<<gE, 256, 0, stream>>>(ei, dinv, E);
  k_dinv<<<gN, 256, 0, stream>>>(dinv, N);

  const float dt = 0.5f;                       // 1 / N_STEPS
  const float wAcc[4]  = {1.f, 2.f, 2.f, 1.f}; // RK4 accumulation weights
  const float cStep[4] = {0.5f * dt, 0.5f * dt, dt, 0.f};  // next-probe coefs

  for (int step = 0; step < 2; ++step) {
    const float* yin = (step == 0) ? x : Yout;
    for (int ev = 0; ev < 4; ++ev) {
      const float* gin = (ev == 0) ? yin : YP;
      k_gemm1<<<gRows, 256, 0, stream>>>(gin, W1, T1, N);
      k_selfloop_init<<<gN, 256, 0, stream>>>(T1, dinv, H, N);
      k_agg_edges<<<gE, 256, 0, stream>>>(ei, dinv, T1, H, E);
      k_relu_bias<<<gN, 256, 0, stream>>>(H, b1, N);
      k_selfloop_init<<<gN, 256, 0, stream>>>(H, dinv, A2, N);
      k_agg_edges<<<gE, 256, 0, stream>>>(ei, dinv, H, A2, E);
      k_gemm2_fused<<<gRows, 256, 0, stream>>>(A2, W2, b2, yin, ACC, YP,
                                               wAcc[ev], cStep[ev],
                                               /*accumulate=*/(ev > 0),
                                               /*writeYp=*/(ev < 3), N);
    }
    k_combine<<<gTot, 256, 0, stream>>>(yin, ACC, Yout, dt / 6.0f, total);
  }
}